// CausalLaplacianLinearAttention_14594298872235
// MI455X (gfx1250) — compile-verified
//
#include <hip/hip_runtime.h>
#include <hip/hip_bf16.h>

// ---------------------------------------------------------------------------
// CausalLaplacianLinearAttention for MI455X (gfx1250, wave32, WMMA).
//
// ~43 of ~46 GFLOP are 16 batched f32 512^3 GEMMs (Newton-Schulz inverse).
// Working set (~76MB) fits in 192MB L2 -> f32 matrix-throughput bound.
// V_WMMA_F32_16X16X4_F32 keeps NS numerically stable (bf16 diverges the
// iterative inverse). GEMM is register-blocked: each wave owns a 2x2 grid of
// 16x16 accumulators so 2 A-frags + 2 B-frags feed 4 WMMAs (1:1 WMMA:DS-load
// ratio). global_prefetch_b8 (via __builtin_prefetch) pulls the next K chunk
// while the current one is computing.
// ---------------------------------------------------------------------------

#define DIM_  512
#define H_    8
#define HD_   64
#define B_    2
#define S_    512
#define BH_   (B_ * H_)
#define LAM_  4.0f
#define EPS_  1e-5f
#define NS_ITERS_ 5

typedef __attribute__((ext_vector_type(2))) float v2f;
typedef __attribute__((ext_vector_type(8))) float v8f;

// ------------------------- generic batched WMMA GEMM -----------------------
// C[z][m][n] = sum_k A[z][m][k]*B[z][k][n]  (+bias[n] | 2I - acc epilogue)
// A elem: Abase + zb*a_bo + zh*a_bi + m*sa_r + k*sa_c
// B elem: Bbase + zb*b_bo + zh*b_bi + k*sb_r + n*sb_c
// C elem: Cbase + zb*c_bo + zh*c_bi + m*ldc  + n      (zb=z/zdiv, zh=z%zdiv)
//
// 8 waves arranged 2 (M) x 4 (N); each wave computes WTM x WTN 16x16 tiles.
// Block tile: BM = 32*WTM, BN = 64*WTN. All problem dims divide evenly.
#define TK 32

template <int WTM, int WTN>
__launch_bounds__(256)
__global__ void wmma_gemm_f32_kernel(
    const float* __restrict__ A, const float* __restrict__ B,
    const float* __restrict__ bias, float* __restrict__ C,
    int K,
    long sa_r, long sa_c, long sb_r, long sb_c, long ldc,
    long a_bo, long a_bi, long b_bo, long b_bi, long c_bo, long c_bi,
    int zdiv, int mode)
{
    constexpr int BM = 2 * 16 * WTM;
    constexpr int BN = 4 * 16 * WTN;
    __shared__ float As[BM][TK + 1];
    __shared__ float Bs[BN][TK + 1];   // [n][k]: both fragments are b64 DS loads

    const int z  = blockIdx.z;
    const int zb = z / zdiv;
    const int zh = z % zdiv;
    A += (long)zb * a_bo + (long)zh * a_bi;
    B += (long)zb * b_bo + (long)zh * b_bi;
    C += (long)zb * c_bo + (long)zh * c_bi;

    const int m0   = blockIdx.y * BM;
    const int n0   = blockIdx.x * BN;
    const int tid  = threadIdx.x;
    const int wave = tid >> 5;
    const int lane = tid & 31;
    const int wm   = wave >> 2;        // 0..1
    const int wn   = wave & 3;         // 0..3

    v8f acc[WTM][WTN];
    #pragma unroll
    for (int im = 0; im < WTM; ++im)
        #pragma unroll
        for (int in = 0; in < WTN; ++in)
            acc[im][in] = (v8f){0.f,0.f,0.f,0.f,0.f,0.f,0.f,0.f};

    const int amBase = wm * 16 * WTM + (lane & 15);
    const int bnBase = wn * 16 * WTN + (lane & 15);
    const int kk0    = (lane >> 4) * 2;

    for (int k0 = 0; k0 < K; k0 += TK) {
        // stage A tile: BM*TK elems
        #pragma unroll
        for (int idx = tid; idx < BM * TK; idx += 256) {
            int m = idx / TK, k = idx % TK;
            As[m][k] = A[(long)(m0 + m) * sa_r + (long)(k0 + k) * sa_c];
        }
        // stage B tile (transposed into LDS): BN*TK elems
        #pragma unroll
        for (int idx = tid; idx < BN * TK; idx += 256) {
            int n = idx / TK, k = idx % TK;
            Bs[n][k] = B[(long)(k0 + k) * sb_r + (long)(n0 + n) * sb_c];
        }
        // prefetch next K chunk into L2/L0 while we compute this one
        if (k0 + TK < K) {
            int ma = tid / TK, ka = tid % TK;
            __builtin_prefetch(&A[(long)(m0 + ma) * sa_r + (long)(k0 + TK + ka) * sa_c], 0, 1);
            int nb = tid / TK, kb = tid % TK;
            __builtin_prefetch(&B[(long)(k0 + TK + kb) * sb_r + (long)(n0 + nb) * sb_c], 0, 1);
        }
        __syncthreads();

        #pragma unroll
        for (int kk = 0; kk < TK; kk += 4) {
            v2f a[WTM], b[WTN];
            #pragma unroll
            for (int im = 0; im < WTM; ++im) {
                a[im].x = As[amBase + im * 16][kk + kk0];
                a[im].y = As[amBase + im * 16][kk + kk0 + 1];
            }
            #pragma unroll
            for (int in = 0; in < WTN; ++in) {
                b[in].x = Bs[bnBase + in * 16][kk + kk0];
                b[in].y = Bs[bnBase + in * 16][kk + kk0 + 1];
            }
            #pragma unroll
            for (int im = 0; im < WTM; ++im)
                #pragma unroll
                for (int in = 0; in < WTN; ++in)
                    acc[im][in] = __builtin_amdgcn_wmma_f32_16x16x4_f32(
                        false, a[im], false, b[in],
                        (short)0, acc[im][in], false, false);
        }
        __syncthreads();
    }

    // store (ISA 7.12.2 C layout: lanes 0-15 -> M=r, lanes 16-31 -> M=r+8)
    const int rsel = (lane >> 4) ? 8 : 0;
    #pragma unroll
    for (int im = 0; im < WTM; ++im) {
        const int rbase = m0 + wm * 16 * WTM + im * 16 + rsel;
        #pragma unroll
        for (int in = 0; in < WTN; ++in) {
            const int col = n0 + wn * 16 * WTN + in * 16 + (lane & 15);
            const float bv = (bias != nullptr) ? bias[col] : 0.0f;
            #pragma unroll
            for (int r = 0; r < 8; ++r) {
                int row = rbase + r;
                float v = acc[im][in][r];
                if (mode == 1) v = (row == col) ? (2.0f - v) : (-v);  // 2I - A@X
                else           v += bv;
                C[(long)row * ldc + col] = v;
            }
        }
    }
}

// --------------------- Laplacian kernel: W = exp(-|q-k|_1/lam)*tril --------
__launch_bounds__(256)
__global__ void lap_kernel(const float* __restrict__ qkv, float* __restrict__ W)
{
    __shared__ float qs[16][HD_ + 1];
    __shared__ float ks[16][HD_ + 1];

    const int z = blockIdx.z;            // b*H + h
    const int b = z / H_, h = z % H_;
    const int i0 = blockIdx.y * 16, j0 = blockIdx.x * 16;
    const int tid = threadIdx.x;

    // qkv layout [B,S,3,H,HD]
    for (int idx = tid; idx < 16 * HD_; idx += 256) {
        int r = idx >> 6, d = idx & 63;
        qs[r][d] = qkv[((long)(b * S_ + i0 + r) * (3 * H_) + h) * HD_ + d];
        ks[r][d] = qkv[((long)(b * S_ + j0 + r) * (3 * H_) + H_ + h) * HD_ + d];
    }
    __syncthreads();

    const int tx = tid & 15, ty = tid >> 4;
    const int i = i0 + ty, j = j0 + tx;
    float s = 0.0f;
    #pragma unroll
    for (int d = 0; d < HD_; ++d) s += fabsf(qs[ty][d] - ks[tx][d]);
    float w = (j <= i) ? __expf(-s * (1.0f / LAM_)) : 0.0f;
    W[((long)z * S_ + i) * S_ + j] = w;
}

// ------------- NS scale: 1 / (max col sum * max row sum), per batch --------
__launch_bounds__(256)
__global__ void ns_scale_kernel(const float* __restrict__ W, float* __restrict__ scale)
{
    __shared__ float red[256];
    const int z = blockIdx.x;
    const float* Aa = W + (long)z * S_ * S_;
    const int t = threadIdx.x;

    float cmax = 0.0f;
    for (int j = t; j < S_; j += 256) {
        float s = 0.0f;
        for (int i = 0; i < S_; ++i) s += fabsf(Aa[(long)i * S_ + j]);
        cmax = fmaxf(cmax, s);
    }
    float rmax = 0.0f;
    for (int i = t; i < S_; i += 256) {
        float s = 0.0f;
        for (int j = 0; j < S_; ++j) s += fabsf(Aa[(long)i * S_ + j]);
        rmax = fmaxf(rmax, s);
    }

    red[t] = cmax; __syncthreads();
    for (int s = 128; s > 0; s >>= 1) {
        if (t < s) red[t] = fmaxf(red[t], red[t + s]);
        __syncthreads();
    }
    float n1 = red[0]; __syncthreads();

    red[t] = rmax; __syncthreads();
    for (int s = 128; s > 0; s >>= 1) {
        if (t < s) red[t] = fmaxf(red[t], red[t + s]);
        __syncthreads();
    }
    float ninf = red[0];

    if (t == 0) scale[z] = 1.0f / (n1 * ninf);
}

// ----------------- X0 = A^T * scale  (16x16 LDS transpose tiles) -----------
__launch_bounds__(256)
__global__ void transpose_scale_kernel(const float* __restrict__ W,
                                       const float* __restrict__ scale,
                                       float* __restrict__ X)
{
    __shared__ float tbuf[16][17];
    const int z = blockIdx.z;
    const long base = (long)z * S_ * S_;
    const int i0 = blockIdx.y * 16, j0 = blockIdx.x * 16;
    const int tx = threadIdx.x & 15, ty = threadIdx.x >> 4;

    tbuf[ty][tx] = W[base + (long)(j0 + ty) * S_ + (i0 + tx)];
    __syncthreads();
    X[base + (long)(i0 + ty) * S_ + (j0 + tx)] = tbuf[tx][ty] * scale[z];
}

// --------- whitening stats over (batch, query) dims: per (h, key j) --------
__launch_bounds__(256)
__global__ void stats_kernel(const float* __restrict__ W,
                             float* __restrict__ mu, float* __restrict__ rs)
{
    const int idx = blockIdx.x * 256 + threadIdx.x;  // h*S + j
    const int h = idx / S_, j = idx % S_;
    float sum = 0.0f, sq = 0.0f;
    for (int b = 0; b < B_; ++b) {
        const float* base = W + (long)(b * H_ + h) * S_ * S_ + j;
        for (int i = 0; i < S_; ++i) {
            float v = base[(long)i * S_];
            sum += v; sq += v * v;
        }
    }
    const float n = (float)(B_ * S_);
    float m = sum / n;
    float var = (sq - n * m * m) / (n - 1.0f);   // unbiased (ddof=1)
    mu[idx] = m;
    rs[idx] = rsqrtf(var + EPS_);
}

// ----------------------- W_norm = (W - mu) * rsigma ------------------------
__launch_bounds__(256)
__global__ void wnorm_kernel(const float* __restrict__ W,
                             const float* __restrict__ mu,
                             const float* __restrict__ rs,
                             float* __restrict__ Wn)
{
    long idx = (long)blockIdx.x * 256 + threadIdx.x;  // over B*H*S*S
    int j = (int)(idx % S_);
    long t = idx / S_;
    int z = (int)(t / S_);
    int h = z % H_;
    Wn[idx] = (W[idx] - mu[h * S_ + j]) * rs[h * S_ + j];
}

// ---------------------------------------------------------------------------
extern "C" void kernel_launch(void* const* d_in, const int* in_sizes, int n_in,
                              void* d_out, int out_size, void* d_ws, size_t ws_size,
                              hipStream_t stream)
{
    (void)in_sizes; (void)n_in; (void)out_size; (void)ws_size;
    const float* x      = (const float*)d_in[0];
    const float* qkv_w  = (const float*)d_in[1];
    const float* qkv_b  = (const float*)d_in[2];
    const float* proj_w = (const float*)d_in[3];
    const float* proj_b = (const float*)d_in[4];
    float* out = (float*)d_out;

    // ---- workspace carve-out ----
    char* ws = (char*)d_ws;
    size_t off = 0;
    auto alloc = [&](size_t nfloats) {
        float* p = (float*)(ws + off);
        off += ((nfloats * sizeof(float) + 255) / 256) * 256;
        return p;
    };
    float* qkv   = alloc((size_t)B_ * S_ * 3 * DIM_);       //  6 MB [B,S,3,H,HD]
    float* W     = alloc((size_t)BH_ * S_ * S_);            // 16 MB
    float* X0    = alloc((size_t)BH_ * S_ * S_);            // 16 MB
    float* X1    = alloc((size_t)BH_ * S_ * S_);            // 16 MB
    float* T     = alloc((size_t)BH_ * S_ * S_);            // 16 MB (NS temp, later W_norm)
    float* ctx1  = alloc((size_t)BH_ * S_ * HD_);           //  2 MB
    float* ctx2  = alloc((size_t)BH_ * S_ * HD_);           //  2 MB
    float* ga    = alloc((size_t)B_ * S_ * DIM_);           //  2 MB
    float* scale = alloc(BH_);
    float* mu    = alloc((size_t)H_ * S_);
    float* rs    = alloc((size_t)H_ * S_);

    const dim3 blk(256);
    const long SS = (long)S_ * S_;

    // big-tile config: BM=64, BN=128 ; skinny config (N=64): BM=64, BN=64
    constexpr int BMb = 64, BNb = 128;
    constexpr int BMs = 64, BNs = 64;

    // 1) qkv = x @ qkv_w^T + qkv_b : [1024 x 1536], K=512
    wmma_gemm_f32_kernel<2, 2><<<dim3(3 * DIM_ / BNb, B_ * S_ / BMb, 1), blk, 0, stream>>>(
        x, qkv_w, qkv_b, qkv, DIM_,
        /*A*/ DIM_, 1, /*B (b[k][n]=qkv_w[n*DIM+k])*/ 1, DIM_, /*ldc*/ 3 * DIM_,
        0, 0, 0, 0, 0, 0, 1, /*mode*/ 0);

    // 2) W = exp(-|q-k|_1 / lam) * tril
    lap_kernel<<<dim3(S_ / 16, S_ / 16, BH_), blk, 0, stream>>>(qkv, W);

    // 3) Newton-Schulz init: scale + X0 = W^T * scale
    ns_scale_kernel<<<dim3(BH_), blk, 0, stream>>>(W, scale);
    transpose_scale_kernel<<<dim3(S_ / 16, S_ / 16, BH_), blk, 0, stream>>>(W, scale, X0);

    // 4) 5 NS iterations: T = 2I - W@X ; Xn = X@T
    float* Xc = X0;
    float* Xn = X1;
    for (int it = 0; it < NS_ITERS_; ++it) {
        wmma_gemm_f32_kernel<2, 2><<<dim3(S_ / BNb, S_ / BMb, BH_), blk, 0, stream>>>(
            W, Xc, nullptr, T, S_,
            S_, 1, S_, 1, S_,
            SS, 0, SS, 0, SS, 0, 1, /*mode 2I-AX*/ 1);
        wmma_gemm_f32_kernel<2, 2><<<dim3(S_ / BNb, S_ / BMb, BH_), blk, 0, stream>>>(
            Xc, T, nullptr, Xn, S_,
            S_, 1, S_, 1, S_,
            SS, 0, SS, 0, SS, 0, 1, 0);
        float* tmp = Xc; Xc = Xn; Xn = tmp;
    }
    // final inverse lives in Xc

    // 5) whitening stats over (b, i) per (h, j)
    stats_kernel<<<dim3(H_ * S_ / 256), blk, 0, stream>>>(W, mu, rs);

    // 6) ctx1 = W^T @ v    (A = W with swapped strides; B = v inside qkv, zdiv=H)
    wmma_gemm_f32_kernel<2, 1><<<dim3(HD_ / BNs, S_ / BMs, BH_), blk, 0, stream>>>(
        W, qkv + 2 * H_ * HD_, nullptr, ctx1, S_,
        /*A = W^T*/ 1, S_, /*B = v*/ 3 * DIM_, 1, /*ldc*/ HD_,
        /*a_bo,a_bi*/ (long)H_ * SS, SS,
        /*b_bo,b_bi*/ (long)S_ * 3 * DIM_, HD_,
        /*c_bo,c_bi*/ (long)H_ * S_ * HD_, (long)S_ * HD_,
        /*zdiv*/ H_, 0);

    // 7) W_norm (reuse T, free after NS loop)
    wnorm_kernel<<<dim3((unsigned)((long)BH_ * SS / 256)), blk, 0, stream>>>(W, mu, rs, T);

    // 8) ctx2 = W_inv @ ctx1
    wmma_gemm_f32_kernel<2, 1><<<dim3(HD_ / BNs, S_ / BMs, BH_), blk, 0, stream>>>(
        Xc, ctx1, nullptr, ctx2, S_,
        S_, 1, HD_, 1, HD_,
        SS, 0, (long)S_ * HD_, 0, (long)S_ * HD_, 0, 1, 0);

    // 9) ga = W_norm @ ctx2, scattered into [B,S,DIM] layout (head-interleaved)
    wmma_gemm_f32_kernel<2, 1><<<dim3(HD_ / BNs, S_ / BMs, BH_), blk, 0, stream>>>(
        T, ctx2, nullptr, ga, S_,
        S_, 1, HD_, 1, /*ldc*/ DIM_,
        /*a*/ (long)H_ * SS, SS,
        /*b*/ (long)H_ * S_ * HD_, (long)S_ * HD_,
        /*c*/ (long)S_ * DIM_, HD_,
        H_, 0);

    // 10) out = ga @ proj_w^T + proj_b : [1024 x 512], K=512
    wmma_gemm_f32_kernel<2, 2><<<dim3(DIM_ / BNb, B_ * S_ / BMb, 1), blk, 0, stream>>>(
        ga, proj_w, proj_b, out, DIM_,
        DIM_, 1, 1, DIM_, DIM_,
        0, 0, 0, 0, 0, 0, 1, 0);
}